// anfis_model_38027640438905
// MI455X (gfx1250) — compile-verified
//
#include <hip/hip_runtime.h>

#define NUM_VAR 8
#define BATCH   2048
#define TOTAL   6561
#define NVP1    9      // NUM_VAR + 1 (bias)

typedef __attribute__((ext_vector_type(2))) float v2f;
typedef __attribute__((ext_vector_type(8))) float v8f;

// ---------------------------------------------------------------------------
// Kernel 1: Cv[v] = sum_j conseq[j*9 + v], v = 0..8   ->  ws[0..8]
// 236 KB read by one workgroup, coalesced, with gfx1250 global_prefetch.
// ---------------------------------------------------------------------------
__global__ void k_reduce_conseq(const float* __restrict__ cp, float* __restrict__ ws) {
    __shared__ float sm[NVP1][256];
    float acc[NVP1];
#pragma unroll
    for (int v = 0; v < NVP1; ++v) acc[v] = 0.0f;

    for (int j = threadIdx.x; j < TOTAL; j += 256) {
        const float* row = cp + j * NVP1;
        __builtin_prefetch(row + 256 * NVP1, 0, 0);   // global_prefetch_b8
#pragma unroll
        for (int v = 0; v < NVP1; ++v) acc[v] += row[v];
    }
#pragma unroll
    for (int v = 0; v < NVP1; ++v) sm[v][threadIdx.x] = acc[v];
    __syncthreads();
    for (int off = 128; off > 0; off >>= 1) {
        if ((int)threadIdx.x < off) {
#pragma unroll
            for (int v = 0; v < NVP1; ++v)
                sm[v][threadIdx.x] += sm[v][threadIdx.x + off];
        }
        __syncthreads();
    }
    if (threadIdx.x == 0) {
#pragma unroll
        for (int v = 0; v < NVP1; ++v) ws[v] = sm[v][0];
    }
}

// ---------------------------------------------------------------------------
// Kernel 2: per batch element b: generalized-bell memberships.
//   F_k(b) = prod_i 1/(1 + ((x_i - c)/a)^2 ^ bexp),  k = 0..2
// Stores F0(b) into d_out (scratch, overwritten by kernel 3), and block-
// reduces (F0+F1+F2) into partials[blockIdx] (8 blocks -> ws[9..16]).
// Global normalizer S = 2187 * sum_b (F0+F1+F2)(b)  [jnp.sum is a full sum].
// ---------------------------------------------------------------------------
__global__ void k_fire(const float* __restrict__ input,  // (8, 2048)
                       const float* __restrict__ pp,     // (24, 3)
                       float* __restrict__ f0_out,       // = d_out (scratch)
                       float* __restrict__ partials) {   // ws + 9
    const int b = blockIdx.x * 256 + threadIdx.x;
    float xin[NUM_VAR];
#pragma unroll
    for (int v = 0; v < NUM_VAR; ++v) xin[v] = input[v * BATCH + b];

    float F0 = 1.0f, F1 = 1.0f, F2 = 1.0f;
#pragma unroll
    for (int i = 0; i < NUM_VAR; ++i) {
#pragma unroll
        for (int k = 0; k < 3; ++k) {
            const int s = 3 * i + k;
            const float a  = pp[3 * s + 0];
            const float be = pp[3 * s + 1];
            const float c  = pp[3 * s + 2];
            const float t  = (xin[i] - c) / a;
            const float m  = 1.0f / (1.0f + __powf(t * t, be));
            if (k == 0)      F0 *= m;
            else if (k == 1) F1 *= m;
            else             F2 *= m;
        }
    }
    f0_out[b] = F0;

    __shared__ float sm[256];
    sm[threadIdx.x] = F0 + F1 + F2;
    __syncthreads();
    for (int off = 128; off > 0; off >>= 1) {
        if ((int)threadIdx.x < off) sm[threadIdx.x] += sm[threadIdx.x + off];
        __syncthreads();
    }
    if (threadIdx.x == 0) partials[blockIdx.x] = sm[0];
}

// ---------------------------------------------------------------------------
// Kernel 3: out[b] = (F0(b)/S) * ( sum_{v<8} input[v,b]*Cv[v] + Cv[8] )
// The 9-term dot for 16 batch elements is computed with two chained
// V_WMMA_F32_16X16X4_F32 (K = 0..3 and K = 4..7), bias folded into C.
//   A (16x4 f32, ISA layout): lanes 0-15 hold K0/K1, lanes 16-31 hold K2/K3.
//   B (4x16): columns replicated with Cv -> every column of D is the dot.
//   D: lane 0 VGPRs 0..7 = rows 0..7, lane 16 VGPRs 0..7 = rows 8..15.
// EXEC is all-ones through both WMMAs (selects only, no branches).
// ---------------------------------------------------------------------------
__global__ void k_output(const float* __restrict__ input,  // (8, 2048)
                         const float* __restrict__ ws,     // [0..8]=Cv, [9..16]=partials
                         float* __restrict__ out) {        // reads F0, writes result
    const int lane = threadIdx.x & 31;
    const bool hi  = lane >= 16;
    const int wave = (blockIdx.x * blockDim.x + threadIdx.x) >> 5;
    const int base = wave * 16;                 // 16 batch elements per wave
    const int e    = base + (lane & 15);

    // Uniform scalars (s_load): Cv and the global normalizer.
    float Cv[NVP1];
#pragma unroll
    for (int v = 0; v < NVP1; ++v) Cv[v] = ws[v];
    float S = 0.0f;
#pragma unroll
    for (int p = 0; p < 8; ++p) S += ws[9 + p];
    const float inv_s = 1.0f / (2187.0f * S);

    const float g = out[e] * inv_s;             // F0(e) / S  (out = scratch F0)

    // A tiles: this lane supplies K = {0,1} (lo half) or {2,3} (hi half).
    const int k0 = hi ? 2 : 0;
    v2f a0, a1, b0, b1;
    a0.x = input[(k0 + 0) * BATCH + e];
    a0.y = input[(k0 + 1) * BATCH + e];
    a1.x = input[(k0 + 4) * BATCH + e];
    a1.y = input[(k0 + 5) * BATCH + e];
    b0.x = hi ? Cv[2] : Cv[0];
    b0.y = hi ? Cv[3] : Cv[1];
    b1.x = hi ? Cv[6] : Cv[4];
    b1.y = hi ? Cv[7] : Cv[5];

    v8f acc;
#pragma unroll
    for (int p = 0; p < 8; ++p) acc[p] = Cv[8];  // bias * C8 in every cell

    acc = __builtin_amdgcn_wmma_f32_16x16x4_f32(false, a0, false, b0,
                                                (short)0, acc, false, false);
    acc = __builtin_amdgcn_wmma_f32_16x16x4_f32(false, a1, false, b1,
                                                (short)0, acc, false, false);

    // acc[p] = dot for element base + p + (hi ? 8 : 0); scale by that
    // element's g (held by lane p or 8+p) and store from lanes 0 and 16.
    float res[8];
#pragma unroll
    for (int p = 0; p < 8; ++p) {
        const float gp = __shfl(g, (hi ? 8 : 0) + p, 32);
        res[p] = gp * acc[p];
    }
    if ((lane & 15) == 0) {
        const int o = base + (hi ? 8 : 0);
#pragma unroll
        for (int p = 0; p < 8; ++p) out[o + p] = res[p];
    }
}

// ---------------------------------------------------------------------------
extern "C" void kernel_launch(void* const* d_in, const int* in_sizes, int n_in,
                              void* d_out, int out_size, void* d_ws, size_t ws_size,
                              hipStream_t stream) {
    const float* input = (const float*)d_in[0];   // (8, 2048) f32
    const float* pp    = (const float*)d_in[1];   // (24, 3)  f32
    const float* cp    = (const float*)d_in[2];   // (59049, 1) f32
    float* out = (float*)d_out;                   // (2048, 1) f32
    float* ws  = (float*)d_ws;                    // uses 17 floats

    k_reduce_conseq<<<1, 256, 0, stream>>>(cp, ws);          // ws[0..8]  = Cv
    k_fire<<<BATCH / 256, 256, 0, stream>>>(input, pp, out, ws + 9); // F0 + partials
    k_output<<<BATCH / (16 * 8), 256, 0, stream>>>(input, ws, out);  // 128 waves
}